// LossTDSurv_90795608637484
// MI455X (gfx1250) — compile-verified
//
#include <hip/hip_runtime.h>
#include <stdint.h>

// ---------------------------------------------------------------------------
// LogisticHazard survival loss, MI455X (gfx1250).
//
// Memory-bound: 134 MiB read -> 1 float.  Floor ~5.8us @ 23.3 TB/s.
// Math is collapsed to one logf(1-h) per element:
//   S   = sum_{t<=idx} log(1-h_t)
//   cond_sum        = S - log(1-h_idx) - log(1-h_{idx-1})
//   sum_{t<=idx}bce = -S - e*phi_idx          (softplus(phi)= -log(1-h+eps))
// Data movement uses the CDNA5 async path: each wave double-buffers one
// 512B row-pair in LDS via global_load_async_to_lds_b128 (1 instr / 512B),
// synchronized with s_wait_asynccnt (in-order completion per wave).
// ---------------------------------------------------------------------------

#define NBLOCKS  1024
#define NTHREADS 256
#define NWAVES   (NTHREADS / 32)
#define T_BINS   64

__global__ __launch_bounds__(NTHREADS)
void surv_loss_main(const float* __restrict__ preds,
                    const float* __restrict__ target,
                    double* __restrict__ partials,
                    unsigned long long nPairs)
{
    __shared__ alignas(16) float  tile[NWAVES][2][2 * T_BINS]; // 8 KB
    __shared__ double             wsum[4][NWAVES];

    const int lane = threadIdx.x & 31;
    const int wave = threadIdx.x >> 5;
    const int half = lane >> 4;     // which row of the pair this lane reduces
    const int lh   = lane & 15;     // 4-element group within the row

    const unsigned long long waveId = (unsigned long long)blockIdx.x * NWAVES + wave;
    const unsigned long long stride = (unsigned long long)NBLOCKS * NWAVES;

    // Generic pointers to LDS have the LDS byte offset in their low 32 bits
    // (flat LDS aperture).
    const uint32_t lds0 = (uint32_t)(uintptr_t)&tile[wave][0][lane * 4];
    const uint32_t lds1 = (uint32_t)(uintptr_t)&tile[wave][1][lane * 4];

    float lz = 0.f, es = 0.f, lc = 0.f, nl = 0.f;

    unsigned long long p = waveId;
    int cur = 0;
    if (p < nPairs) {
        const float* g = preds + p * (2ull * T_BINS) + (unsigned)(lane * 4);
        asm volatile("global_load_async_to_lds_b128 %0, %1, off"
                     :: "v"(lds0), "v"(g) : "memory");
    }

    for (; p < nPairs; p += stride) {
        const unsigned long long pn = p + stride;
        if (pn < nPairs) {
            const float* g = preds + pn * (2ull * T_BINS) + (unsigned)(lane * 4);
            const uint32_t nxt = (cur == 0) ? lds1 : lds0;
            // prior iteration's ds reads of this buffer must drain first
            asm volatile("s_wait_dscnt 0x0" ::: "memory");
            asm volatile("global_load_async_to_lds_b128 %0, %1, off"
                         :: "v"(nxt), "v"(g) : "memory");
            // async loads complete in order: <=1 pending => current buf ready
            asm volatile("s_wait_asynccnt 0x1" ::: "memory");
        } else {
            asm volatile("s_wait_asynccnt 0x0" ::: "memory");
        }

        const unsigned long long row = 2ull * p + (unsigned)half;
        const float idxf = target[3ull * row + 0];
        const float ev   = target[3ull * row + 1];
        const int   idx  = (int)idxf;

        const float4 h4 = reinterpret_cast<const float4*>(&tile[wave][cur][0])[lane];
        const float hv[4] = {h4.x, h4.y, h4.z, h4.w};

        float S = 0.f, a = 0.f, a1 = 0.f, hl = 0.f;
        #pragma unroll
        for (int k = 0; k < 4; ++k) {
            const int   t   = lh * 4 + k;
            const float h   = hv[k];
            const float l1m = logf(1.0f - h);
            S += (t <= idx) ? l1m : 0.0f;
            if (t == idx)     { a  = l1m; hl = h; }
            if (t == idx - 1) { a1 = l1m; }
        }
        // butterfly over each 16-lane half (masks < 16 stay within the half)
        #pragma unroll
        for (int m = 1; m < 16; m <<= 1) {
            S  += __shfl_xor(S,  m);
            a  += __shfl_xor(a,  m);
            a1 += __shfl_xor(a1, m);
            hl += __shfl_xor(hl, m);
        }
        if (lh == 0) {
            const float cond   = S - a - a1;                 // sum_{t<=idx-2} log(1-h)
            const float log_hl = logf(hl);
            const float phi    = logf(hl + 1e-12f) - logf(1.0f - hl + 1e-12f);
            const float wt     = logf(fmaxf(1.0f - expf(cond), 1e-8f));
            lz += ev * (-(log_hl + cond));                   // L_z numerator
            es += ev;                                        // L_z denominator
            lc += (1.0f - ev) * cond + ev * wt;              // -L_c * B
            nl += -S - ev * phi;                             // nll * B
        }
        cur ^= 1;
    }

    // fold lane16 (row r+1) into lane0 (row r)
    lz += __shfl_xor(lz, 16);
    es += __shfl_xor(es, 16);
    lc += __shfl_xor(lc, 16);
    nl += __shfl_xor(nl, 16);
    if (lane == 0) {
        wsum[0][wave] = (double)lz;
        wsum[1][wave] = (double)es;
        wsum[2][wave] = (double)lc;
        wsum[3][wave] = (double)nl;
    }
    __syncthreads();
    if (threadIdx.x == 0) {
        double s0 = 0, s1 = 0, s2 = 0, s3 = 0;
        for (int w = 0; w < NWAVES; ++w) {
            s0 += wsum[0][w]; s1 += wsum[1][w];
            s2 += wsum[2][w]; s3 += wsum[3][w];
        }
        partials[blockIdx.x * 4 + 0] = s0;
        partials[blockIdx.x * 4 + 1] = s1;
        partials[blockIdx.x * 4 + 2] = s2;
        partials[blockIdx.x * 4 + 3] = s3;
    }
}

__global__ __launch_bounds__(256)
void surv_loss_final(const double* __restrict__ partials,
                     float* __restrict__ out, double Bd)
{
    __shared__ double red[4][256];
    double s[4] = {0, 0, 0, 0};
    for (int i = threadIdx.x; i < NBLOCKS; i += 256) {
        #pragma unroll
        for (int q = 0; q < 4; ++q) s[q] += partials[i * 4 + q];
    }
    #pragma unroll
    for (int q = 0; q < 4; ++q) red[q][threadIdx.x] = s[q];
    __syncthreads();
    for (int off = 128; off > 0; off >>= 1) {
        if ((int)threadIdx.x < off) {
            #pragma unroll
            for (int q = 0; q < 4; ++q)
                red[q][threadIdx.x] += red[q][threadIdx.x + off];
        }
        __syncthreads();
    }
    if (threadIdx.x == 0) {
        const double lz = red[0][0], es = red[1][0];
        const double lc = red[2][0], nl = red[3][0];
        // alpha=0.5, beta=1:  0.5*L_z + 0.5*(-(lc/B)) + nll/B
        const double res = 0.5 * (lz / es) - 0.5 * (lc / Bd) + nl / Bd;
        out[0] = (float)res;
    }
}

extern "C" void kernel_launch(void* const* d_in, const int* in_sizes, int n_in,
                              void* d_out, int out_size, void* d_ws, size_t ws_size,
                              hipStream_t stream)
{
    (void)n_in; (void)out_size; (void)ws_size;
    const float* preds  = (const float*)d_in[0];
    const float* target = (const float*)d_in[1];
    const long long B = in_sizes[1] / 3;                 // target is (B,3)
    const unsigned long long nPairs = (unsigned long long)(B / 2);
    double* partials = (double*)d_ws;                    // 1024*4 doubles = 32 KB

    surv_loss_main<<<NBLOCKS, NTHREADS, 0, stream>>>(preds, target, partials, nPairs);
    surv_loss_final<<<1, 256, 0, stream>>>(partials, (float*)d_out, (double)B);
}